// Model1_38448547233947
// MI455X (gfx1250) — compile-verified
//
#include <hip/hip_runtime.h>
#include <math.h>

// ---------------------------------------------------------------------------
// Quantum-conv forward for MI455X (gfx1250, wave32).
// Circuit params are patch-independent -> whole circuit == one 16x16 complex
// unitary U. Then  Y = U * V  (V: 16 x 2500 real Kron-product amplitudes) is
// two fp32 GEMMs mapped to V_WMMA_F32_16X16X4_F32 (16x16 D tile = 16 patches).
// ---------------------------------------------------------------------------

#define IMG     100
#define OUTD    50
#define NPATCH  2500
#define NTILES  157            // ceil(2500 / 16)

// float offsets into workspace
#define UR_OFF   0             // 16x16 Re(U), row-major [row][col]
#define UI_OFF   256           // 16x16 Im(U)
#define PART_OFF 512           // NTILES * 8 per-tile logit partials

typedef __attribute__((ext_vector_type(2))) float v2f;
typedef __attribute__((ext_vector_type(8))) float v8f;

// ---------------------------------------------------------------------------
// Kernel 1: build the full-circuit 16x16 unitary. One wave; lane j < 16 owns
// column j of U (16 complex values in registers). Gate loops inside a block
// are unrolled (constant state indices -> registers); the 5-block outer loop
// is NOT unrolled to keep code size small (I-fetch matters more than the
// handful of dynamic param loads).
// ---------------------------------------------------------------------------
__global__ __launch_bounds__(32) void build_unitary_kernel(
    const float* __restrict__ u3p, const float* __restrict__ cu3p,
    float* __restrict__ ws) {
  const int j = threadIdx.x;
  if (j >= 16) return;

  float sr[16], si[16];
#pragma unroll
  for (int k = 0; k < 16; ++k) { sr[k] = (k == j) ? 1.f : 0.f; si[k] = 0.f; }

  const int pc[4] = {0, 1, 2, 3};   // CU3 control wires
  const int pt[4] = {1, 2, 3, 0};   // CU3 target wires

#pragma unroll 1
  for (int blk = 0; blk < 5; ++blk) {
    // ---- four single-qubit U3 gates ----
#pragma unroll
    for (int w = 0; w < 4; ++w) {
      const int base = (blk * 4 + w) * 3;
      const float th = u3p[base], ph = u3p[base + 1], la = u3p[base + 2];
      float ct, sn, cph, sph, cla, sla, cpl, spl;
      sincosf(0.5f * th, &sn, &ct);
      sincosf(ph, &sph, &cph);
      sincosf(la, &sla, &cla);
      sincosf(ph + la, &spl, &cpl);
      const float u00r = ct,        u00i = 0.f;
      const float u01r = -cla * sn, u01i = -sla * sn;
      const float u10r =  cph * sn, u10i =  sph * sn;
      const float u11r =  cpl * ct, u11i =  spl * ct;
      const int m = 8 >> w;                 // wire w -> bit (3-w)
#pragma unroll
      for (int k = 0; k < 16; ++k) {
        if (k & m) continue;
        const int k2 = k | m;
        const float ar = sr[k],  ai = si[k];
        const float br = sr[k2], bi = si[k2];
        sr[k]  = u00r * ar - u00i * ai + u01r * br - u01i * bi;
        si[k]  = u00r * ai + u00i * ar + u01r * bi + u01i * br;
        sr[k2] = u10r * ar - u10i * ai + u11r * br - u11i * bi;
        si[k2] = u10r * ai + u10i * ar + u11r * bi + u11i * br;
      }
    }
    // ---- four controlled-U3 gates (apply u on target where ctrl bit = 1) --
#pragma unroll
    for (int g = 0; g < 4; ++g) {
      const int base = (blk * 4 + g) * 3;
      const float th = cu3p[base], ph = cu3p[base + 1], la = cu3p[base + 2];
      float ct, sn, cph, sph, cla, sla, cpl, spl;
      sincosf(0.5f * th, &sn, &ct);
      sincosf(ph, &sph, &cph);
      sincosf(la, &sla, &cla);
      sincosf(ph + la, &spl, &cpl);
      const float u00r = ct,        u00i = 0.f;
      const float u01r = -cla * sn, u01i = -sla * sn;
      const float u10r =  cph * sn, u10i =  sph * sn;
      const float u11r =  cpl * ct, u11i =  spl * ct;
      const int mc = 8 >> pc[g], mt = 8 >> pt[g];
#pragma unroll
      for (int k = 0; k < 16; ++k) {
        if (!(k & mc) || (k & mt)) continue;
        const int k2 = k | mt;
        const float ar = sr[k],  ai = si[k];
        const float br = sr[k2], bi = si[k2];
        sr[k]  = u00r * ar - u00i * ai + u01r * br - u01i * bi;
        si[k]  = u00r * ai + u00i * ar + u01r * bi + u01i * br;
        sr[k2] = u10r * ar - u10i * ai + u11r * br - u11i * bi;
        si[k2] = u10r * ai + u10i * ar + u11r * bi + u11i * br;
      }
    }
  }
#pragma unroll
  for (int k = 0; k < 16; ++k) {
    ws[UR_OFF + k * 16 + j] = sr[k];
    ws[UI_OFF + k * 16 + j] = si[k];
  }
}

// ---------------------------------------------------------------------------
// Kernel 2: one wave per tile of 16 patches. A = Re/Im(U) slabs (16x4),
// B = amplitude matrix V (4x16) built in registers, D accumulated over 4
// K-chunks with V_WMMA_F32_16X16X4_F32. Then probs -> <Z_w> -> logit partials.
//
// Layouts (per CDNA5 ISA):
//   A 16x4 f32 : lane%16 = M; lanes<16 hold K={0,1}, lanes>=16 hold K={2,3}
//   D 16x16 f32: lane%16 = N; VGPR r = row r (lanes<16) / row r+8 (lanes>=16)
//   B 4x16 f32 : assumed symmetric to A: lane%16 = N; half selects K pair
// ---------------------------------------------------------------------------
__global__ __launch_bounds__(256) void qconv_wmma_kernel(
    const float* __restrict__ x, const float* __restrict__ W,
    float* __restrict__ ws) {
  const int lane = threadIdx.x & 31;
  const int tile = blockIdx.x * (blockDim.x >> 5) + (threadIdx.x >> 5);
  if (tile >= NTILES) return;            // wave-uniform: EXEC stays all-ones

  const int  pl    = lane & 15;          // patch column within tile (= N)
  const int  p0    = tile * 16 + pl;
  const bool valid = (p0 < NPATCH);
  const int  p     = valid ? p0 : (NPATCH - 1);
  const int  prow  = p / OUTD, pcol = p % OUTD;

  // patch pixels (2x2, stride 2), wires 0..3 row-major within patch.
  // args are in [0, 0.5) -> native v_sin/v_cos are plenty accurate.
  const float* img = x + (2 * prow) * IMG + 2 * pcol;
  const float x0 = img[0], x1 = img[1], x2 = img[IMG], x3 = img[IMG + 1];
  const float c0 = __cosf(0.5f * x0), s0 = __sinf(0.5f * x0);
  const float c1 = __cosf(0.5f * x1), s1 = __sinf(0.5f * x1);
  const float c2 = __cosf(0.5f * x2), s2 = __sinf(0.5f * x2);
  const float c3 = __cosf(0.5f * x3), s3 = __sinf(0.5f * x3);

  const bool hi = (lane >= 16);
  const int  kb = hi ? 2 : 0;            // K-pair base for this lane half

  const float* Ur = ws + UR_OFF;
  const float* Ui = ws + UI_OFF;

  v8f dr = {0.f, 0.f, 0.f, 0.f, 0.f, 0.f, 0.f, 0.f};
  v8f di = {0.f, 0.f, 0.f, 0.f, 0.f, 0.f, 0.f, 0.f};

#pragma unroll
  for (int k0 = 0; k0 < 16; k0 += 4) {
    const int ka = k0 + kb;
    v2f a_r, a_i, bv;
    a_r.x = Ur[pl * 16 + ka];     a_r.y = Ur[pl * 16 + ka + 1];
    a_i.x = Ui[pl * 16 + ka];     a_i.y = Ui[pl * 16 + ka + 1];
    // V[k][p] = prod_q (bit_{3-q}(k) ? sin : cos)
    {
      const int k = ka;
      bv.x = ((k & 8) ? s0 : c0) * ((k & 4) ? s1 : c1) *
             ((k & 2) ? s2 : c2) * ((k & 1) ? s3 : c3);
    }
    {
      const int k = ka + 1;
      bv.y = ((k & 8) ? s0 : c0) * ((k & 4) ? s1 : c1) *
             ((k & 2) ? s2 : c2) * ((k & 1) ? s3 : c3);
    }
    dr = __builtin_amdgcn_wmma_f32_16x16x4_f32(false, a_r, false, bv,
                                               (short)0, dr, false, false);
    di = __builtin_amdgcn_wmma_f32_16x16x4_f32(false, a_i, false, bv,
                                               (short)0, di, false, false);
  }

  // probs -> <Z_w> partials for this lane's 8 rows
  float ez[4] = {0.f, 0.f, 0.f, 0.f};
  const int rbase = hi ? 8 : 0;
#pragma unroll
  for (int r = 0; r < 8; ++r) {
    const float pr = dr[r] * dr[r] + di[r] * di[r];
    const int   k  = r + rbase;
#pragma unroll
    for (int w = 0; w < 4; ++w)
      ez[w] += ((k >> (3 - w)) & 1) ? -pr : pr;
  }
  // combine the two lane halves (rows 0-7 + rows 8-15 of the same patch)
#pragma unroll
  for (int w = 0; w < 4; ++w)
    ez[w] += __shfl_xor(ez[w], 16, 32);
#pragma unroll
  for (int w = 0; w < 4; ++w)
    if (!valid) ez[w] = 0.f;

  // logits partial: feat index f = pcol*50 + prow (transpose(2,1,0) in ref)
  const int   f  = pcol * OUTD + prow;
  const float w0 = W[f], w1 = W[NPATCH + f];
  float acc[8];
#pragma unroll
  for (int q = 0; q < 4; ++q) {
    acc[q * 2 + 0] = ez[q] * w0;
    acc[q * 2 + 1] = ez[q] * w1;
  }
  // butterfly-reduce across the 16 patches of this wave
#pragma unroll
  for (int m = 1; m <= 8; m <<= 1)
#pragma unroll
    for (int i = 0; i < 8; ++i)
      acc[i] += __shfl_xor(acc[i], m, 32);

  if (lane == 0) {
#pragma unroll
    for (int i = 0; i < 8; ++i)
      ws[PART_OFF + tile * 8 + i] = acc[i];
  }
}

// ---------------------------------------------------------------------------
// Kernel 3: deterministic fixed-order reduction of per-tile partials,
// add bias, stable log-softmax over the 2 classes, write [4,2] output.
// ---------------------------------------------------------------------------
__global__ __launch_bounds__(32) void finalize_kernel(
    const float* __restrict__ ws, const float* __restrict__ b,
    float* __restrict__ out) {
  const int q = threadIdx.x;
  if (q >= 4) return;
  float l0 = b[0], l1 = b[1];
  for (int t = 0; t < NTILES; ++t) {
    l0 += ws[PART_OFF + t * 8 + q * 2 + 0];
    l1 += ws[PART_OFF + t * 8 + q * 2 + 1];
  }
  const float m   = fmaxf(l0, l1);
  const float lse = m + logf(expf(l0 - m) + expf(l1 - m));
  out[q * 2 + 0] = l0 - lse;
  out[q * 2 + 1] = l1 - lse;
}

// ---------------------------------------------------------------------------
extern "C" void kernel_launch(void* const* d_in, const int* in_sizes, int n_in,
                              void* d_out, int out_size, void* d_ws,
                              size_t ws_size, hipStream_t stream) {
  const float* x    = (const float*)d_in[0];   // [1,100,100]
  const float* u3p  = (const float*)d_in[1];   // [5,4,3]
  const float* cu3p = (const float*)d_in[2];   // [5,4,3]
  const float* W    = (const float*)d_in[3];   // [2,2500]
  const float* b    = (const float*)d_in[4];   // [2]
  float* out = (float*)d_out;                  // [4,2] log-softmax
  float* ws  = (float*)d_ws;

  build_unitary_kernel<<<1, 32, 0, stream>>>(u3p, cu3p, ws);

  const int waves_per_block = 256 / 32;
  const int nblocks = (NTILES + waves_per_block - 1) / waves_per_block; // 20
  qconv_wmma_kernel<<<nblocks, 256, 0, stream>>>(x, W, ws);

  finalize_kernel<<<1, 32, 0, stream>>>(ws, b, out);
}